// SGD_21122649162367
// MI455X (gfx1250) — compile-verified
//
#include <hip/hip_runtime.h>
#include <cmath>

#define DIMC   768
#define ATTRC  312
#define HEADSC 4
#define DHC    192
#define BC     32
#define NFC    1024
#define NMC    256
#define BKT    16
#define LPAD   20   // LDS row stride (floats): 80B -> even-k pairs 8B-aligned, float4 rows 16B-aligned

typedef __attribute__((ext_vector_type(2))) float v2f;
typedef __attribute__((ext_vector_type(8))) float v8f;

__device__ __forceinline__ float gelu_erf(float x) {
  return 0.5f * x * (1.0f + erff(x * 0.70710678118654752f));
}

__device__ __forceinline__ float block_reduce_sum(float v, float* red) {
  int tid = threadIdx.x;
  red[tid] = v; __syncthreads();
  for (int o = 128; o > 0; o >>= 1) { if (tid < o) red[tid] += red[tid + o]; __syncthreads(); }
  float r = red[0]; __syncthreads();
  return r;
}

__device__ __forceinline__ float block_reduce_max(float v, float* red) {
  int tid = threadIdx.x;
  red[tid] = v; __syncthreads();
  for (int o = 128; o > 0; o >>= 1) { if (tid < o) red[tid] = fmaxf(red[tid], red[tid + o]); __syncthreads(); }
  float r = red[0]; __syncthreads();
  return r;
}

// load an even-k fragment pair from a padded LDS row as one ds_load_b64
__device__ __forceinline__ v2f lds_pair(const float* p) {
  return *reinterpret_cast<const v2f*>(p);
}

// -------- row normalization: out = (x-mean)/sqrt(biased_var+eps) [*g + b] ------
__global__ __launch_bounds__(256)
void rownorm(const float* __restrict__ x, float* __restrict__ out, int Dd,
             const float* __restrict__ g, const float* __restrict__ gb, float eps) {
  __shared__ float red[256];
  size_t base = (size_t)blockIdx.x * Dd;
  int tid = threadIdx.x;
  float s = 0.f, sq = 0.f;
  for (int i = tid; i < Dd; i += 256) { float v = x[base + i]; s += v; sq += v * v; }
  s  = block_reduce_sum(s,  red);
  sq = block_reduce_sum(sq, red);
  float mean = s / (float)Dd;
  float var  = sq / (float)Dd - mean * mean;
  float inv  = rsqrtf(var + eps);
  for (int i = tid; i < Dd; i += 256) {
    float v = (x[base + i] - mean) * inv;
    if (g) v = v * g[i] + gb[i];
    out[base + i] = v;
  }
}

// -------- general GEMM: out = act( (A .* gain + gbias) @ W + bias + res ) -----
// A:[M,Kd] row-major (gain/gbias applied along K), W:[Kd,N] row-major.
// res_mode: 0 none, 1 res[row*N+col], 2 res[(row%res_mod)*N+col]. act: 0 none,1 gelu.
// Out-of-range A rows are CLAMPED (not zeroed): each output row depends only on
// its own A row, and rows >= M are never stored.
template <bool HAS_GAIN>
__global__ __launch_bounds__(256)
void gemm_wmma_f32(const float* __restrict__ Ax, const float* __restrict__ gain,
                   const float* __restrict__ gbias,
                   const float* __restrict__ W, const float* __restrict__ bias,
                   const float* __restrict__ res, int res_mode, int res_mod,
                   float* __restrict__ out, int M, int N, int Kd, int act) {
  __shared__ float As[64][LPAD];   // As[row][k]
  __shared__ float Bt[64][LPAD];   // Bt[col][k]  (B transposed in LDS)
  int tid = threadIdx.x, lane = tid & 31, wave = tid >> 5;
  int wm = wave & 3, wn = wave >> 2;
  int tileM = blockIdx.y * 64, tileN = blockIdx.x * 64;
  int ar = tid >> 2, ac4 = (tid & 3) << 2;          // A staging coords
  int arow_g = tileM + ar; if (arow_g >= M) arow_g = M - 1;   // clamp
  v8f acc0 = {}; v8f acc1 = {};
  for (int k0 = 0; k0 < Kd; k0 += BKT) {
    {                                               // A tile 64x16: b128 load + b128 LDS store
      float4 av = *reinterpret_cast<const float4*>(&Ax[(size_t)arow_g * Kd + k0 + ac4]);
      if (HAS_GAIN) {
        av.x = av.x * gain[k0 + ac4 + 0] + gbias[k0 + ac4 + 0];
        av.y = av.y * gain[k0 + ac4 + 1] + gbias[k0 + ac4 + 1];
        av.z = av.z * gain[k0 + ac4 + 2] + gbias[k0 + ac4 + 2];
        av.w = av.w * gain[k0 + ac4 + 3] + gbias[k0 + ac4 + 3];
      }
      *reinterpret_cast<float4*>(&As[ar][ac4]) = av;
    }
#pragma unroll
    for (int i = 0; i < 4; ++i) {                   // B tile 16x64 -> transposed store
      int idx = tid + i * 256, r = idx >> 6, c = idx & 63;
      Bt[c][r] = W[(size_t)(k0 + r) * N + tileN + c];
    }
    if (k0 + BKT < Kd) {                            // prefetch next k-tile
      __builtin_prefetch(&Ax[(size_t)arow_g * Kd + k0 + BKT + ac4], 0, 3);
      __builtin_prefetch(&W[(size_t)(k0 + BKT + (tid >> 6)) * N + tileN + (tid & 63)], 0, 3);
    }
    __syncthreads();
    int m = lane & 15, ko = (lane >> 4) << 1;
#pragma unroll
    for (int kk = 0; kk < BKT; kk += 4) {
      v2f a  = lds_pair(&As[wm * 16 + m][kk + ko]);
      v2f b0 = lds_pair(&Bt[wn * 32 + m][kk + ko]);
      v2f b1 = lds_pair(&Bt[wn * 32 + 16 + m][kk + ko]);
      acc0 = __builtin_amdgcn_wmma_f32_16x16x4_f32(false, a, false, b0, (short)0, acc0, false, false);
      acc1 = __builtin_amdgcn_wmma_f32_16x16x4_f32(false, a, false, b1, (short)0, acc1, false, false);
    }
    __syncthreads();
  }
  int rbase = (lane >> 4) * 8, nl = lane & 15;
#pragma unroll
  for (int r = 0; r < 8; ++r) {
    int row = tileM + wm * 16 + rbase + r;
    if (row >= M) continue;
#pragma unroll
    for (int j = 0; j < 2; ++j) {
      int col = tileN + wn * 32 + j * 16 + nl;
      float vl = (j == 0) ? acc0[r] : acc1[r];
      if (bias) vl += bias[col];
      if (res_mode == 1)      vl += res[(size_t)row * N + col];
      else if (res_mode == 2) vl += res[(size_t)(row % res_mod) * N + col];
      if (act == 1) vl = gelu_erf(vl);
      out[(size_t)row * N + col] = vl;
    }
  }
}

// -------- logits = scale * q @ k^T  per (b,h). out row stride Nk ---------------
__global__ __launch_bounds__(256)
void logits_wmma(const float* __restrict__ q, const float* __restrict__ kbuf,
                 float* __restrict__ out, int Nk, float scale) {
  __shared__ float As[64][LPAD];
  __shared__ float Bt[64][LPAD];   // Bt[n][d] = k[tileN+n][k0+d]
  int tid = threadIdx.x, lane = tid & 31, wave = tid >> 5;
  int wm = wave & 3, wn = wave >> 2;
  int b = blockIdx.z >> 2, h = blockIdx.z & 3;
  int tileM = blockIdx.y * 64, tileN = blockIdx.x * 64;
  const float* qh = q + h * DHC;
  const float* kh = kbuf + (size_t)b * Nk * DIMC + h * DHC;
  float* orow = out + (size_t)blockIdx.z * ATTRC * Nk;
  int ar = tid >> 2, ac4 = (tid & 3) << 2;
  int arow_g = tileM + ar; if (arow_g >= ATTRC) arow_g = ATTRC - 1;  // clamp
  v8f acc0 = {}; v8f acc1 = {};
  for (int k0 = 0; k0 < DHC; k0 += BKT) {
    {                                               // q tile: b128 both sides
      float4 av = *reinterpret_cast<const float4*>(&qh[(size_t)arow_g * DIMC + k0 + ac4]);
      *reinterpret_cast<float4*>(&As[ar][ac4]) = av;
    }
    {                                               // k tile: contiguous global + LDS
      float4 bv = *reinterpret_cast<const float4*>(&kh[(size_t)(tileN + ar) * DIMC + k0 + ac4]);
      *reinterpret_cast<float4*>(&Bt[ar][ac4]) = bv;
    }
    if (k0 + BKT < DHC) {
      __builtin_prefetch(&qh[(size_t)arow_g * DIMC + k0 + BKT + ac4], 0, 3);
      __builtin_prefetch(&kh[(size_t)(tileN + ar) * DIMC + k0 + BKT + ac4], 0, 3);
    }
    __syncthreads();
    int m = lane & 15, ko = (lane >> 4) << 1;
#pragma unroll
    for (int kk = 0; kk < BKT; kk += 4) {
      v2f a  = lds_pair(&As[wm * 16 + m][kk + ko]);
      v2f b0 = lds_pair(&Bt[wn * 32 + m][kk + ko]);
      v2f b1 = lds_pair(&Bt[wn * 32 + 16 + m][kk + ko]);
      acc0 = __builtin_amdgcn_wmma_f32_16x16x4_f32(false, a, false, b0, (short)0, acc0, false, false);
      acc1 = __builtin_amdgcn_wmma_f32_16x16x4_f32(false, a, false, b1, (short)0, acc1, false, false);
    }
    __syncthreads();
  }
  int rbase = (lane >> 4) * 8, nl = lane & 15;
#pragma unroll
  for (int r = 0; r < 8; ++r) {
    int row = tileM + wm * 16 + rbase + r;
    if (row >= ATTRC) continue;
#pragma unroll
    for (int j = 0; j < 2; ++j) {
      int col = tileN + wn * 32 + j * 16 + nl;
      orow[(size_t)row * Nk + col] = ((j == 0) ? acc0[r] : acc1[r]) * scale;
    }
  }
}

// -------- ctx[b, :, h*192 + :] = attn(b,h) @ v(b)[:, h*192 + :] ---------------
__global__ __launch_bounds__(256)
void attnv_wmma(const float* __restrict__ attn, const float* __restrict__ vbuf,
                float* __restrict__ ctx, int Nk) {
  __shared__ float As[64][LPAD];
  __shared__ float Bt[64][LPAD];   // Bt[n][kk] = v[k0+kk][tileN+n]
  int tid = threadIdx.x, lane = tid & 31, wave = tid >> 5;
  int wm = wave & 3, wn = wave >> 2;
  int b = blockIdx.z >> 2, h = blockIdx.z & 3;
  int tileM = blockIdx.y * 64, tileN = blockIdx.x * 64;  // tileN over Dh
  const float* arow = attn + (size_t)blockIdx.z * ATTRC * Nk;
  const float* vh = vbuf + (size_t)b * Nk * DIMC + h * DHC;
  float* crow = ctx + (size_t)b * ATTRC * DIMC + h * DHC;
  int ar = tid >> 2, ac4 = (tid & 3) << 2;
  int arow_g = tileM + ar; if (arow_g >= ATTRC) arow_g = ATTRC - 1;  // clamp
  v8f acc0 = {}; v8f acc1 = {};
  for (int k0 = 0; k0 < Nk; k0 += BKT) {
    {                                               // attn tile: b128 both sides
      float4 av = *reinterpret_cast<const float4*>(&arow[(size_t)arow_g * Nk + k0 + ac4]);
      *reinterpret_cast<float4*>(&As[ar][ac4]) = av;
    }
#pragma unroll
    for (int i = 0; i < 4; ++i) {                   // v tile: coalesced global, transposed LDS
      int idx = tid + i * 256, r = idx >> 6, c = idx & 63;
      Bt[c][r] = vh[(size_t)(k0 + r) * DIMC + tileN + c];
    }
    if (k0 + BKT < Nk) {
      __builtin_prefetch(&arow[(size_t)arow_g * Nk + k0 + BKT + ac4], 0, 3);
      __builtin_prefetch(&vh[(size_t)(k0 + BKT + (tid >> 6)) * DIMC + tileN + (tid & 63)], 0, 3);
    }
    __syncthreads();
    int m = lane & 15, ko = (lane >> 4) << 1;
#pragma unroll
    for (int kk = 0; kk < BKT; kk += 4) {
      v2f a  = lds_pair(&As[wm * 16 + m][kk + ko]);
      v2f b0 = lds_pair(&Bt[wn * 32 + m][kk + ko]);
      v2f b1 = lds_pair(&Bt[wn * 32 + 16 + m][kk + ko]);
      acc0 = __builtin_amdgcn_wmma_f32_16x16x4_f32(false, a, false, b0, (short)0, acc0, false, false);
      acc1 = __builtin_amdgcn_wmma_f32_16x16x4_f32(false, a, false, b1, (short)0, acc1, false, false);
    }
    __syncthreads();
  }
  int rbase = (lane >> 4) * 8, nl = lane & 15;
#pragma unroll
  for (int r = 0; r < 8; ++r) {
    int row = tileM + wm * 16 + rbase + r;
    if (row >= ATTRC) continue;
#pragma unroll
    for (int j = 0; j < 2; ++j) {
      int col = tileN + wn * 32 + j * 16 + nl;
      crow[(size_t)row * DIMC + col] = (j == 0) ? acc0[r] : acc1[r];
    }
  }
}

// -------- in-place top-K + tempered softmax scatter per logits row ------------
__global__ __launch_bounds__(256)
void topk_softmax(float* __restrict__ logits, int Nk,
                  const int* __restrict__ topk_p, const float* __restrict__ tau_p) {
  __shared__ float vals[1024];
  __shared__ float red[256];
  int tid = threadIdx.x;
  float* row = logits + (size_t)blockIdx.x * Nk;
  int K = *topk_p; if (K > Nk) K = Nk;
  for (int i = tid; i < Nk; i += 256) vals[i] = row[i];
  __syncthreads();
  auto key = [](float f) {
    unsigned u = __float_as_uint(f);
    return (u & 0x80000000u) ? ~u : (u | 0x80000000u);
  };
  // radix bisection: largest t with count(key >= t) >= K  -> t is K-th largest key
  unsigned t = 0u;
  for (int bit = 31; bit >= 0; --bit) {
    unsigned cand = t | (1u << bit);
    float c = 0.f;
    for (int i = tid; i < Nk; i += 256) if (key(vals[i]) >= cand) c += 1.f;
    float total = block_reduce_sum(c, red);
    if (total >= (float)K) t = cand;
  }
  float cg = 0.f;
  for (int i = tid; i < Nk; i += 256) if (key(vals[i]) > t) cg += 1.f;
  int G = (int)block_reduce_sum(cg, red);
  int E = K - G;  // #ties at t to keep (first occurrences, like stable top_k)
  float mx = -3.402823466e38f;
  for (int i = tid; i < Nk; i += 256) mx = fmaxf(mx, vals[i]);
  float vmax = block_reduce_max(mx, red);
  float tau = fmaxf(*tau_p, 1e-8f);
  float itau = 1.0f / tau;
  float s = 0.f;
  for (int i = tid; i < Nk; i += 256) {
    unsigned u = key(vals[i]);
    bool sel = (u > t);
    if (!sel && u == t) {
      int rank = 0;
      for (int j = 0; j < i; ++j) if (key(vals[j]) == t) ++rank;
      sel = (rank < E);
    }
    if (sel) s += expf((vals[i] - vmax) * itau);
  }
  float sum = block_reduce_sum(s, red);
  float inv = 1.0f / sum;
  for (int i = tid; i < Nk; i += 256) {
    unsigned u = key(vals[i]);
    bool sel = (u > t);
    if (!sel && u == t) {
      int rank = 0;
      for (int j = 0; j < i; ++j) if (key(vals[j]) == t) ++rank;
      sel = (rank < E);
    }
    row[i] = sel ? expf((vals[i] - vmax) * itau) * inv : 0.f;
  }
}

// -------- small VALU GEMM for tiny matrices -----------------------------------
// element A[m][k] = Am[m*a_rs + k*a_cs]; W:[K,N] row-major. act: 0 none,1 gelu,2 sigmoid
__global__ __launch_bounds__(256)
void small_mm(const float* __restrict__ Am, int a_rs, int a_cs,
              const float* __restrict__ W, const float* __restrict__ bias,
              float* __restrict__ out, int M, int N, int K, int act) {
  int idx = blockIdx.x * 256 + threadIdx.x;
  if (idx >= M * N) return;
  int m = idx / N, n = idx % N;
  float s = bias ? bias[n] : 0.f;
  for (int k = 0; k < K; ++k) s += Am[(size_t)m * a_rs + (size_t)k * a_cs] * W[(size_t)k * N + n];
  if (act == 1) s = gelu_erf(s);
  else if (act == 2) s = 1.0f / (1.0f + expf(-s));
  out[idx] = s;
}

// -------- adafusion final blend ------------------------------------------------
__global__ __launch_bounds__(256)
void fuse_kernel(const float* __restrict__ xf, const float* __restrict__ xm,
                 const float* __restrict__ alphaV, const float* __restrict__ betaV,
                 float* __restrict__ outF, float* __restrict__ alphaO, float* __restrict__ betaO) {
  __shared__ float red[256];
  int row = blockIdx.x;
  int a = row % ATTRC, b = row / ATTRC;
  int tid = threadIdx.x;
  size_t base = (size_t)row * DIMC;
  float sf = 0.f, sqf = 0.f, smm = 0.f, sqm = 0.f;
  for (int i = tid; i < DIMC; i += 256) {
    float vf = xf[base + i]; sf += vf; sqf += vf * vf;
    float vm = xm[base + i]; smm += vm; sqm += vm * vm;
  }
  sf  = block_reduce_sum(sf,  red);
  sqf = block_reduce_sum(sqf, red);
  smm = block_reduce_sum(smm, red);
  sqm = block_reduce_sum(sqm, red);
  const float n = (float)DIMC;
  float mf = sf / n, mm = smm / n;
  float vf = (sqf - n * mf * mf) / (n - 1.0f);   // unbiased var (ddof=1)
  float vm = (sqm - n * mm * mm) / (n - 1.0f);
  float invf = 1.0f / (vf + 1e-6f);              // reference divides by (var+eps), no sqrt
  float invm = 1.0f / (vm + 1e-6f);
  float al = alphaV[a];
  for (int i = tid; i < DIMC; i += 256) {
    float nf  = (xf[base + i] - mf) * invf;
    float nm2 = (xm[base + i] - mm) * invm;
    outF[base + i] = al * nf + (1.0f - al) * nm2 + betaV[i];
  }
  if (tid == 0) alphaO[row] = al;
  if (a == 0) for (int i = tid; i < DIMC; i += 256) betaO[(size_t)b * DIMC + i] = betaV[i];
}

// ---- block param dict offsets (insertion order of block_params()) ------------
enum { P_LNQ_G = 0, P_LNQ_B, P_LNK_G, P_LNK_B, P_LNV_G, P_LNV_B,
       P_WQ, P_BQ, P_WK, P_BK_, P_WV, P_BV, P_WP, P_BP, P_TAU,
       P_W1, P_B1, P_W2, P_B2, P_LN_G, P_LN_B };

static void run_block(const float* xtok, int Nk, void* const* d_in, int pb,
                      const float* wv, const float* wvhat,
                      float* bufA, float* kbuf, float* vbuf, float* ctxb,
                      float* localb, float* qbuf,
                      float* logits_out, float* block_out,
                      const int* topk_p, hipStream_t stream) {
  const float* lnq_g = (const float*)d_in[pb + P_LNQ_G];
  const float* lnq_b = (const float*)d_in[pb + P_LNQ_B];
  const float* lnk_g = (const float*)d_in[pb + P_LNK_G];
  const float* lnk_b = (const float*)d_in[pb + P_LNK_B];
  const float* lnv_g = (const float*)d_in[pb + P_LNV_G];
  const float* lnv_b = (const float*)d_in[pb + P_LNV_B];
  const float* Wq = (const float*)d_in[pb + P_WQ];  const float* bq = (const float*)d_in[pb + P_BQ];
  const float* Wk = (const float*)d_in[pb + P_WK];  const float* bk = (const float*)d_in[pb + P_BK_];
  const float* Wv = (const float*)d_in[pb + P_WV];  const float* bv = (const float*)d_in[pb + P_BV];
  const float* Wp = (const float*)d_in[pb + P_WP];  const float* bp = (const float*)d_in[pb + P_BP];
  const float* tau = (const float*)d_in[pb + P_TAU];
  const float* W1 = (const float*)d_in[pb + P_W1];  const float* b1 = (const float*)d_in[pb + P_B1];
  const float* W2 = (const float*)d_in[pb + P_W2];  const float* b2 = (const float*)d_in[pb + P_B2];
  const float* ln_g = (const float*)d_in[pb + P_LN_G];
  const float* ln_b = (const float*)d_in[pb + P_LN_B];

  const int Mtok = BC * Nk;
  const int MA = BC * ATTRC;
  const float scale = 1.0f / sqrtf((float)DHC);

  rownorm<<<Mtok, 256, 0, stream>>>(xtok, bufA, DIMC, nullptr, nullptr, 1e-5f);

  dim3 gq(DIMC / 64, (ATTRC + 63) / 64);
  gemm_wmma_f32<true><<<gq, 256, 0, stream>>>(wvhat, lnq_g, lnq_b, Wq, bq, nullptr, 0, 0,
                                              qbuf, ATTRC, DIMC, DIMC, 0);
  dim3 gkv(DIMC / 64, Mtok / 64);
  gemm_wmma_f32<true><<<gkv, 256, 0, stream>>>(bufA, lnk_g, lnk_b, Wk, bk, nullptr, 0, 0,
                                               kbuf, Mtok, DIMC, DIMC, 0);
  gemm_wmma_f32<true><<<gkv, 256, 0, stream>>>(bufA, lnv_g, lnv_b, Wv, bv, nullptr, 0, 0,
                                               vbuf, Mtok, DIMC, DIMC, 0);

  dim3 gl(Nk / 64, (ATTRC + 63) / 64, BC * HEADSC);
  logits_wmma<<<gl, 256, 0, stream>>>(qbuf, kbuf, logits_out, Nk, scale);

  topk_softmax<<<BC * HEADSC * ATTRC, 256, 0, stream>>>(logits_out, Nk, topk_p, tau);

  dim3 gc(DHC / 64, (ATTRC + 63) / 64, BC * HEADSC);
  attnv_wmma<<<gc, 256, 0, stream>>>(logits_out, vbuf, ctxb, Nk);

  // refined + wv -> kbuf (k no longer needed)
  dim3 gp(DIMC / 64, MA / 64);
  gemm_wmma_f32<false><<<gp, 256, 0, stream>>>(ctxb, nullptr, nullptr, Wp, bp, wv, 2, ATTRC,
                                               kbuf, MA, DIMC, DIMC, 0);
  rownorm<<<MA, 256, 0, stream>>>(kbuf, localb, DIMC, ln_g, ln_b, 1e-5f);

  // hidden = gelu(local@W1+b1) -> bufA (xhat no longer needed)
  dim3 gh((2 * DIMC) / 64, MA / 64);
  gemm_wmma_f32<false><<<gh, 256, 0, stream>>>(localb, nullptr, nullptr, W1, b1, nullptr, 0, 0,
                                               bufA, MA, 2 * DIMC, DIMC, 1);
  // out = hidden@W2 + b2 + local
  dim3 go(DIMC / 64, MA / 64);
  gemm_wmma_f32<false><<<go, 256, 0, stream>>>(bufA, nullptr, nullptr, W2, b2, localb, 1, 0,
                                               block_out, MA, DIMC, 2 * DIMC, 0);
}

extern "C" void kernel_launch(void* const* d_in, const int* in_sizes, int n_in,
                              void* d_out, int out_size, void* d_ws, size_t ws_size,
                              hipStream_t stream) {
  (void)in_sizes; (void)n_in; (void)out_size; (void)ws_size;
  const float* x_macro = (const float*)d_in[0];
  const float* x_fine  = (const float*)d_in[1];
  const float* w2v     = (const float*)d_in[2];
  const float* Wwe     = (const float*)d_in[3];
  const float* bwe     = (const float*)d_in[4];
  const int FB = 5, CB = 26, ADA = 47;
  const float* Wa1 = (const float*)d_in[ADA + 0];
  const float* Wa2 = (const float*)d_in[ADA + 1];
  const float* Wa3 = (const float*)d_in[ADA + 2];
  const float* Wb1 = (const float*)d_in[ADA + 3];
  const float* Wb2 = (const float*)d_in[ADA + 4];
  const float* Wb3 = (const float*)d_in[ADA + 5];
  const int* tkf = (const int*)d_in[53];
  const int* tkm = (const int*)d_in[54];

  float* out = (float*)d_out;
  const size_t S1 = (size_t)BC * ATTRC * DIMC;
  float* out_attr  = out;
  float* out_fine  = out + S1;
  float* out_macro = out + 2 * S1;
  float* fine_w    = out + 3 * S1;
  float* macro_w   = fine_w + (size_t)BC * HEADSC * ATTRC * NFC;
  float* alpha_o   = macro_w + (size_t)BC * HEADSC * ATTRC * NMC;
  float* beta_o    = alpha_o + (size_t)BC * ATTRC;

  float* ws = (float*)d_ws;
  size_t o = 0;
  float* bufA   = ws + o; o += (size_t)BC * NFC * DIMC;   // xhat, later MLP hidden
  float* kbuf   = ws + o; o += (size_t)BC * NFC * DIMC;   // k, later refined+wv
  float* vbuf   = ws + o; o += (size_t)BC * NFC * DIMC;
  float* ctxb   = ws + o; o += (size_t)BC * ATTRC * DIMC;
  float* localb = ws + o; o += (size_t)BC * ATTRC * DIMC;
  float* wv     = ws + o; o += (size_t)ATTRC * DIMC;
  float* wvhat  = ws + o; o += (size_t)ATTRC * DIMC;
  float* qbuf   = ws + o; o += (size_t)ATTRC * DIMC;
  float* afeat1 = ws + o; o += (size_t)ATTRC * (DIMC / 2);
  float* afeat2 = ws + o; o += (size_t)ATTRC * (DIMC / 4);
  float* bfeat1 = ws + o; o += (size_t)DIMC * (ATTRC / 2);
  float* bfeat2 = ws + o; o += (size_t)DIMC * (ATTRC / 4);
  float* alphaV = ws + o; o += ATTRC;
  float* betaV  = ws + o; o += DIMC;

  // wv = w2v @ Wwe + bwe  (batch-invariant since w2v has B=1), then normalize
  {
    int total = ATTRC * DIMC;
    small_mm<<<(total + 255) / 256, 256, 0, stream>>>(w2v, 300, 1, Wwe, bwe, wv,
                                                      ATTRC, DIMC, 300, 0);
    rownorm<<<ATTRC, 256, 0, stream>>>(wv, wvhat, DIMC, nullptr, nullptr, 1e-5f);
  }

  run_block(x_fine,  NFC, d_in, FB, wv, wvhat, bufA, kbuf, vbuf, ctxb, localb, qbuf,
            fine_w,  out_fine,  tkf, stream);
  run_block(x_macro, NMC, d_in, CB, wv, wvhat, bufA, kbuf, vbuf, ctxb, localb, qbuf,
            macro_w, out_macro, tkm, stream);

  // adafusion (alpha/beta are batch-invariant as well)
  small_mm<<<(ATTRC * 384 + 255) / 256, 256, 0, stream>>>(wv, DIMC, 1, Wa1, nullptr,
                                                          afeat1, ATTRC, 384, DIMC, 1);
  small_mm<<<(ATTRC * 192 + 255) / 256, 256, 0, stream>>>(afeat1, 384, 1, Wa2, nullptr,
                                                          afeat2, ATTRC, 192, 384, 1);
  small_mm<<<(ATTRC + 255) / 256, 256, 0, stream>>>(afeat2, 192, 1, Wa3, nullptr,
                                                    alphaV, ATTRC, 1, 192, 2);
  small_mm<<<(DIMC * 156 + 255) / 256, 256, 0, stream>>>(wv, 1, DIMC, Wb1, nullptr,
                                                         bfeat1, DIMC, 156, ATTRC, 1);
  small_mm<<<(DIMC * 78 + 255) / 256, 256, 0, stream>>>(bfeat1, 156, 1, Wb2, nullptr,
                                                        bfeat2, DIMC, 78, 156, 1);
  small_mm<<<(DIMC + 255) / 256, 256, 0, stream>>>(bfeat2, 78, 1, Wb3, nullptr,
                                                   betaV, DIMC, 1, 78, 0);

  fuse_kernel<<<BC * ATTRC, 256, 0, stream>>>(out_fine, out_macro, alphaV, betaV,
                                              out_attr, alpha_o, beta_o);
}